// SingleColumnModel_43971875176990
// MI455X (gfx1250) — compile-verified
//
#include <hip/hip_runtime.h>
#include <stdint.h>

#define NZ 256
#define CPL 8          // cells per lane (32 lanes * 8 = 256)
#define WPB 8          // waves per block -> 256 threads

#define DTf       30.0f
#define NSTEPS    16
#define FCORf     1.0e-4f
#define USTR_SFCf 8.0e-5f
#define RFLXf     3.7e-5f
#define TFLX_SFCf (-1.0e-5f)
#define GAMMAf    0.55f

#if defined(__HIP_DEVICE_COMPILE__) && __has_builtin(__builtin_amdgcn_global_load_async_to_lds_b128)
#define HAVE_ASYNC_LDS 1
#else
#define HAVE_ASYNC_LDS 0
#endif

struct Op {
  float Minv[CPL];   // 1/m_i
  float alpha[CPL];  // -a_i/m_i   (fwd recurrence coefficient)
  float cp[CPL];     // c_i/m_i    (bwd recurrence coefficient is -cp)
  float p[CPL];      // local prefix products of alpha
  float pb[CPL];     // local suffix products of -cp
  float Ms[5];       // cross-lane scan multipliers (fwd), 0 where inactive
  float Msb[5];      // cross-lane scan multipliers (bwd), 0 where inactive
};

__device__ __forceinline__ void load8(const float* __restrict__ p, float v[CPL]) {
  const float4* p4 = reinterpret_cast<const float4*>(p);
  float4 x = p4[0], y = p4[1];
  v[0]=x.x; v[1]=x.y; v[2]=x.z; v[3]=x.w;
  v[4]=y.x; v[5]=y.y; v[6]=y.z; v[7]=y.w;
}

__device__ __forceinline__ void store8(float* __restrict__ p, const float v[CPL]) {
  float4* p4 = reinterpret_cast<float4*>(p);
  p4[0] = make_float4(v[0], v[1], v[2], v[3]);
  p4[1] = make_float4(v[4], v[5], v[6], v[7]);
}

// Build the per-operator factorization data (matrix constant across steps).
__device__ __forceinline__ void make_op(const float* __restrict__ akcol,
                                        const float hz[CPL], float hzm, float hzp,
                                        int lane, Op& op) {
  const int s0 = lane * CPL;
  float akL[CPL];
#pragma unroll
  for (int j = 0; j < CPL; ++j) akL[j] = akcol[s0 + j];
  float ak8 = __shfl_down(akL[0], 1, 32);       // ak[s0+8] from next lane
  if (lane == 31) ak8 = akcol[NZ];              // ak[256]

  float a[CPL], bb[CPL], c[CPL];
#pragma unroll
  for (int j = 0; j < CPL; ++j) {
    float hm  = (j == 0)       ? hzm : hz[j - 1];
    float hp  = (j == CPL - 1) ? hzp : hz[j + 1];
    float akn = (j == CPL - 1) ? ak8 : akL[j + 1];
    float aj = (-2.0f * DTf) * akL[j] / (hm + hz[j]);
    float cj = (-2.0f * DTf) * akn   / (hz[j] + hp);
    if (lane == 0  && j == 0)       aj = 0.0f;  // bottom row
    if (lane == 31 && j == CPL - 1) cj = 0.0f;  // surface row
    a[j] = aj; c[j] = cj; bb[j] = hz[j] - aj - cj;
  }

  // One-time serial Thomas factorization, chained lane-by-lane.
  float chain = 0.0f;
  for (int k = 0; k < 32; ++k) {
    float incoming = __shfl(chain, (k == 0) ? 0 : (k - 1), 32);
    if (lane == k) {
      float cpp = (k == 0) ? 0.0f : incoming;
#pragma unroll
      for (int j = 0; j < CPL; ++j) {
        float m  = bb[j] - a[j] * cpp;
        float mi = 1.0f / m;
        op.Minv[j]  = mi;
        op.cp[j]    = c[j] * mi;
        op.alpha[j] = -a[j] * mi;
        cpp = op.cp[j];
      }
      chain = cpp;
    }
  }

  // Local prefix products (forward sweep); lane 0 chain is identically 0 (alpha[0]=0).
  op.p[0] = op.alpha[0];
#pragma unroll
  for (int j = 1; j < CPL; ++j) op.p[j] = op.alpha[j] * op.p[j - 1];
  float P = op.p[CPL - 1];
#pragma unroll
  for (int s = 0; s < 5; ++s) {
    int d = 1 << s;
    float Pp = __shfl_up(P, d, 32);
    bool act = (lane >= d);
    op.Ms[s] = act ? P : 0.0f;                  // multiplier used on incoming Q
    P = act ? P * Pp : P;
  }

  // Local suffix products (backward sweep); lane 31 chain is identically 0 (cp[255]=0).
  op.pb[CPL - 1] = -op.cp[CPL - 1];
#pragma unroll
  for (int j = CPL - 2; j >= 0; --j) op.pb[j] = -op.cp[j] * op.pb[j + 1];
  float Pb = op.pb[0];
#pragma unroll
  for (int s = 0; s < 5; ++s) {
    int d = 1 << s;
    float Pn = __shfl_down(Pb, d, 32);
    bool act = (lane + d < 32);
    op.Msb[s] = act ? Pb : 0.0f;
    Pb = act ? Pb * Pn : Pb;
  }
}

// Solve two tridiagonal systems sharing one factorization. rhs in, solution out (in place).
__device__ __forceinline__ void solve2(const Op& op, float* __restrict__ x0,
                                       float* __restrict__ x1, int lane) {
  // forward sweep: dp_j = alpha_j*dp_{j-1} + f_j*Minv_j
  x0[0] *= op.Minv[0];
  x1[0] *= op.Minv[0];
#pragma unroll
  for (int j = 1; j < CPL; ++j) {
    x0[j] = fmaf(op.alpha[j], x0[j - 1], x0[j] * op.Minv[j]);
    x1[j] = fmaf(op.alpha[j], x1[j - 1], x1[j] * op.Minv[j]);
  }
  float Q0 = x0[CPL - 1], Q1 = x1[CPL - 1];
#pragma unroll
  for (int s = 0; s < 5; ++s) {
    int d = 1 << s;
    float u0 = __shfl_up(Q0, d, 32);
    float u1 = __shfl_up(Q1, d, 32);
    Q0 = fmaf(op.Ms[s], u0, Q0);               // Ms==0 for inactive lanes
    Q1 = fmaf(op.Ms[s], u1, Q1);
  }
  float c0 = __shfl_up(Q0, 1, 32);
  float c1 = __shfl_up(Q1, 1, 32);
  if (lane == 0) { c0 = 0.0f; c1 = 0.0f; }
#pragma unroll
  for (int j = 0; j < CPL; ++j) {
    x0[j] = fmaf(op.p[j], c0, x0[j]);
    x1[j] = fmaf(op.p[j], c1, x1[j]);
  }
  // backward sweep: x_j = dp_j - cp_j*x_{j+1}
#pragma unroll
  for (int j = CPL - 2; j >= 0; --j) {
    x0[j] = fmaf(-op.cp[j], x0[j + 1], x0[j]);
    x1[j] = fmaf(-op.cp[j], x1[j + 1], x1[j]);
  }
  float Qb0 = x0[0], Qb1 = x1[0];
#pragma unroll
  for (int s = 0; s < 5; ++s) {
    int d = 1 << s;
    float n0 = __shfl_down(Qb0, d, 32);
    float n1 = __shfl_down(Qb1, d, 32);
    Qb0 = fmaf(op.Msb[s], n0, Qb0);
    Qb1 = fmaf(op.Msb[s], n1, Qb1);
  }
  float cb0 = __shfl_down(Qb0, 1, 32);
  float cb1 = __shfl_down(Qb1, 1, 32);
  if (lane == 31) { cb0 = 0.0f; cb1 = 0.0f; }
#pragma unroll
  for (int j = 0; j < CPL; ++j) {
    x0[j] = fmaf(op.pb[j], cb0, x0[j]);
    x1[j] = fmaf(op.pb[j], cb1, x1[j]);
  }
}

__global__ __launch_bounds__(32 * WPB) void scm_wavecol_kernel(
    const float* __restrict__ in_t, const float* __restrict__ in_s,
    const float* __restrict__ in_u, const float* __restrict__ in_v,
    const float* __restrict__ in_hz, const float* __restrict__ in_akt,
    const float* __restrict__ in_akv, float* __restrict__ out, int B) {
  const int lane = threadIdx.x & 31;
  const int wave = threadIdx.x >> 5;
  const int col  = blockIdx.x * WPB + wave;
#if HAVE_ASYNC_LDS
  __shared__ __align__(16) float hz_lds[WPB][NZ];
#endif
  if (col >= B) return;  // uniform per wave

  const int    s0    = lane * CPL;
  const size_t cbase = (size_t)col * NZ;
  const size_t abase = (size_t)col * (NZ + 1);

  // CDNA5 global_prefetch of this wave's column data
  __builtin_prefetch(in_t  + cbase + s0, 0, 1);
  __builtin_prefetch(in_s  + cbase + s0, 0, 1);
  __builtin_prefetch(in_u  + cbase + s0, 0, 1);
  __builtin_prefetch(in_v  + cbase + s0, 0, 1);
  __builtin_prefetch(in_akt + abase + s0, 0, 1);
  __builtin_prefetch(in_akv + abase + s0, 0, 1);

  float hz[CPL];
#if HAVE_ASYNC_LDS
  {
    // Stage hz column into LDS with the CDNA5 async global->LDS path (ASYNCcnt).
    // Builtin signature (from toolchain diagnostic): (int4 AS1*, int4 AS3*, imm offset, imm cpol)
    typedef int v4i __attribute__((vector_size(16)));
    typedef __attribute__((address_space(1))) v4i* gptr_t;
    typedef __attribute__((address_space(3))) v4i* lptr_t;
    gptr_t gsrc = (gptr_t)(uintptr_t)(in_hz + cbase);                       // 16B aligned
    lptr_t ldst = (lptr_t)(uint32_t)(uintptr_t)(&hz_lds[wave][0]);          // LDS byte offset
    __builtin_amdgcn_global_load_async_to_lds_b128(gsrc + lane,      ldst + lane,      0, 0);
    __builtin_amdgcn_global_load_async_to_lds_b128(gsrc + 32 + lane, ldst + 32 + lane, 0, 0);
#if __has_builtin(__builtin_amdgcn_s_wait_asynccnt)
    __builtin_amdgcn_s_wait_asynccnt(0);
#else
    asm volatile("s_wait_asynccnt 0" ::: "memory");
#endif
#pragma unroll
    for (int j = 0; j < CPL; ++j) hz[j] = hz_lds[wave][s0 + j];
  }
#else
  load8(in_hz + cbase + s0, hz);
#endif

  float tv[CPL], sv[CPL], uv[CPL], vv[CPL];
  load8(in_t + cbase + s0, tv);
  load8(in_s + cbase + s0, sv);
  load8(in_u + cbase + s0, uv);
  load8(in_v + cbase + s0, vv);

  const float hzm = __shfl_up(hz[CPL - 1], 1, 32);   // hz[s0-1]
  const float hzp = __shfl_down(hz[0], 1, 32);       // hz[s0+8]

  Op opT, opV;
  make_op(in_akt + abase, hz, hzm, hzp, lane, opT);
  make_op(in_akv + abase, hz, hzm, hzp, lane, opV);

  // Constant solar-flux forcing: g[i] = dt*(fc[i+1]-fc[i]), fc[0]=0, fc[256]=RFLX+TFLX_SFC
  float gflux[CPL];
  {
    float S1[CPL], S2[CPL];
    float p1 = 1.0f, p2 = 1.0f;
#pragma unroll
    for (int j = CPL - 1; j >= 0; --j) {
      float x1 = hz[j] * (-1.0f / 0.35f);
      float x2 = hz[j] * (-1.0f / 23.0f);
      float e1 = (x1 > -20.0f) ? __expf(x1) : 0.0f;
      float e2 = (x2 > -20.0f) ? __expf(x2) : 0.0f;
      p1 *= e1; p2 *= e2;
      S1[j] = p1; S2[j] = p2;                    // suffix product within lane
    }
    float I1 = S1[0], I2 = S2[0];                // cross-lane suffix products
#pragma unroll
    for (int s = 0; s < 5; ++s) {
      int d = 1 << s;
      float n1 = __shfl_down(I1, d, 32);
      float n2 = __shfl_down(I2, d, 32);
      if (lane + d < 32) { I1 *= n1; I2 *= n2; }
    }
    float G1 = __shfl_down(I1, 1, 32);
    float G2 = __shfl_down(I2, 1, 32);
    if (lane == 31) { G1 = 1.0f; G2 = 1.0f; }
    const float r1 = 0.58f, r2 = 0.42f;
#pragma unroll
    for (int j = 0; j < CPL; ++j) {
      float lo1 = S1[j] * G1, lo2 = S2[j] * G2;
      float hi1 = ((j < CPL - 1) ? S1[j + 1] : 1.0f) * G1;
      float hi2 = ((j < CPL - 1) ? S2[j + 1] : 1.0f) * G2;
      float fc_lo = RFLXf * fmaf(r1, lo1, r2 * lo2);
      float fc_hi = RFLXf * fmaf(r1, hi1, r2 * hi2);
      if (lane == 0  && j == 0)       fc_lo = 0.0f;
      if (lane == 31 && j == CPL - 1) fc_hi = RFLXf + TFLX_SFCf;
      gflux[j] = DTf * (fc_hi - fc_lo);
    }
  }

  // Coriolis constants
  const float cff   = (DTf * FCORf) * (DTf * FCORf);
  const float cff1  = 1.0f / (1.0f + GAMMAf * GAMMAf * cff);
  const float A2    = cff1 * (1.0f - GAMMAf * (1.0f - GAMMAf) * cff);
  const float A3    = cff1 * (DTf * FCORf);
  const float sfc_u = DTf * USTR_SFCf;

#pragma unroll 1
  for (int step = 0; step < NSTEPS; ++step) {
    // tracers: f_t = hz*t + g ; f_s = hz*s  (fc_s is identically zero in the reference)
#pragma unroll
    for (int j = 0; j < CPL; ++j) {
      tv[j] = fmaf(hz[j], tv[j], gflux[j]);
      sv[j] = hz[j] * sv[j];
    }
    solve2(opT, tv, sv, lane);

    // momentum: fu = hz*(A2*u + A3*v) ; fv = hz*(A2*v - A3*u) ; surface stress on u
#pragma unroll
    for (int j = 0; j < CPL; ++j) {
      float uu = uv[j], ww = vv[j];
      uv[j] = hz[j] * fmaf(A3, ww, A2 * uu);
      vv[j] = hz[j] * fmaf(-A3, uu, A2 * ww);
    }
    if (lane == 31) uv[CPL - 1] += sfc_u;
    solve2(opV, uv, vv, lane);
  }

  float* ocol = out + (size_t)col * (4 * NZ);
  store8(ocol + 0 * NZ + s0, tv);
  store8(ocol + 1 * NZ + s0, sv);
  store8(ocol + 2 * NZ + s0, uv);
  store8(ocol + 3 * NZ + s0, vv);
}

extern "C" void kernel_launch(void* const* d_in, const int* in_sizes, int n_in,
                              void* d_out, int out_size, void* d_ws, size_t ws_size,
                              hipStream_t stream) {
  (void)n_in; (void)out_size; (void)d_ws; (void)ws_size;
  const int B = in_sizes[0] / NZ;
  dim3 block(32 * WPB);
  dim3 grid((unsigned)((B + WPB - 1) / WPB));
  scm_wavecol_kernel<<<grid, block, 0, stream>>>(
      (const float*)d_in[0], (const float*)d_in[1], (const float*)d_in[2],
      (const float*)d_in[3], (const float*)d_in[4], (const float*)d_in[5],
      (const float*)d_in[6], (float*)d_out, B);
}